// WordSimilarityModel_72619307041261
// MI455X (gfx1250) — compile-verified
//
#include <hip/hip_runtime.h>

typedef float v2f __attribute__((ext_vector_type(2)));
typedef float v4f __attribute__((ext_vector_type(4)));
typedef float v8f __attribute__((ext_vector_type(8)));

#define DIM 128
#define ROWPAD 132            // LDS floats per row: 128 + 4 pad -> conflict-free ds_load_b64
#define TILE_ROWS 16
#define TILE_FLOATS (TILE_ROWS * ROWPAD)   // 2112 floats per tile buffer
#define WAVES_PER_BLOCK 2
#define SIM_BLOCKS 4096
#define TOP_BLOCKS 128
#define EPS 1e-6f

// ---------------------------------------------------------------------------
// CDNA5 async global->LDS helpers (ASYNCcnt-tracked DMA, §15.18 / CDNA5 ISA)
// ---------------------------------------------------------------------------
__device__ __forceinline__ void async_ld_b128(unsigned lds_byte, const void* gaddr) {
    // GV mode: copies 16B per lane from the global address to the LDS byte address in vdst
    asm volatile("global_load_async_to_lds_b128 %0, %1, off"
                 :: "v"(lds_byte), "v"(gaddr)
                 : "memory");
}

#define WAIT_ASYNC(n) asm volatile("s_wait_asynccnt %0" :: "n"(n) : "memory")

// Issue one 16x128 tile (16 rows, one async b128 per lane per row) into LDS.
__device__ __forceinline__ void stage_tile_async(const float* __restrict__ W, size_t rowbase,
                                                 unsigned lds_byte_base, int lane) {
#pragma unroll
    for (int r = 0; r < TILE_ROWS; ++r) {
        async_ld_b128(lds_byte_base + (unsigned)((r * ROWPAD + 4 * lane) * 4),
                      (const void*)(W + rowbase + (size_t)r * DIM + 4 * lane));
    }
}

// ---------------------------------------------------------------------------
// Kernel 1: query norm (single wave)
// ---------------------------------------------------------------------------
__global__ __launch_bounds__(32) void qnorm_kernel(const float* __restrict__ W,
                                                   const int* __restrict__ qidx,
                                                   float* __restrict__ qnorm_out) {
    const int lane = threadIdx.x;
    const float* q = W + (size_t)qidx[0] * DIM;
    float s = 0.f;
#pragma unroll
    for (int i = 0; i < 4; ++i) {
        float v = q[lane + 32 * i];
        s = fmaf(v, v, s);
    }
#pragma unroll
    for (int off = 16; off >= 1; off >>= 1) s += __shfl_xor(s, off, 32);
    if (lane == 0) qnorm_out[0] = sqrtf(s);
}

// ---------------------------------------------------------------------------
// Kernel 2: cosine similarity via V_WMMA_F32_16X16X4_F32 with double-buffered
// async global->LDS staging. Each wave owns two 16-row tile buffers: DMA tile
// t+1 while computing tile t (32 WMMA accumulations over K=128, row norms as
// VALU side product of the same fragments).
// ---------------------------------------------------------------------------
__global__ __launch_bounds__(WAVES_PER_BLOCK * 32) void sim_kernel(
    const float* __restrict__ W, const int* __restrict__ qidx,
    const float* __restrict__ qnorm_p, float* __restrict__ sims, int V) {
    __shared__ float lds[WAVES_PER_BLOCK * 2 * TILE_FLOATS];

    const int lane = threadIdx.x & 31;
    const int wave = threadIdx.x >> 5;
    const int kk   = (lane < 16) ? 0 : 2;   // K sub-offset per ISA 16x4 f32 A layout
    const int row  = lane & 15;             // M = lane % 16

    float* buf0 = lds + (size_t)wave * 2 * TILE_FLOATS;
    float* buf1 = buf0 + TILE_FLOATS;
    // Generic LDS pointer low 32 bits == wave-relative LDS byte address (ISA §10.2)
    const unsigned buf0_byte = (unsigned)(uintptr_t)buf0;
    const unsigned buf1_byte = (unsigned)(uintptr_t)buf1;

    const float* q  = W + (size_t)qidx[0] * DIM;
    const float  qn = qnorm_p[0];

    // B operand (query) fragments, loaded once, live in registers for all tiles.
    // B[k][n] = q[k] for every column n -> all 16 output columns hold the dots.
    v2f qb[32];
#pragma unroll
    for (int kc = 0; kc < 32; ++kc) {
        qb[kc] = *(const v2f*)(q + 4 * kc + kk);
    }

    const int numTiles = V / TILE_ROWS;
    const int stride   = gridDim.x * WAVES_PER_BLOCK;
    int tile = blockIdx.x * WAVES_PER_BLOCK + wave;

    if (tile < numTiles)
        stage_tile_async(W, (size_t)tile * TILE_ROWS * DIM, buf0_byte, lane);

    int cur = 0;
    for (; tile < numTiles; tile += stride, cur ^= 1) {
        const int nxt = tile + stride;
        if (nxt < numTiles) {
            // prefetch next tile into the other buffer, then wait for current
            stage_tile_async(W, (size_t)nxt * TILE_ROWS * DIM,
                             cur ? buf0_byte : buf1_byte, lane);
            WAIT_ASYNC(TILE_ROWS);   // 16 newer transfers may remain in flight
        } else {
            WAIT_ASYNC(0);
        }
        const float* wbuf = cur ? buf1 : buf0;

        v8f c = {};
        float ss = 0.f;
#pragma unroll
        for (int kc = 0; kc < 32; ++kc) {
            const v2f a = *(const v2f*)(wbuf + row * ROWPAD + 4 * kc + kk);
            c = __builtin_amdgcn_wmma_f32_16x16x4_f32(
                false, a, false, qb[kc], (short)0, c, false, false);
            ss = fmaf(a.x, a.x, ss);
            ss = fmaf(a.y, a.y, ss);
        }
        // Complete per-row sum of squares: lane L and L^16 cover a row together.
        const float ssrow = ss + __shfl_xor(ss, 16, 32);

        // C/D layout: lanes 0-15 hold rows 0-7 in VGPRs 0-7; lanes 16-31 rows 8-15.
        const int vb = (lane < 16) ? 0 : 8;
        float s[8];
#pragma unroll
        for (int v = 0; v < 8; ++v) {
            const float wn = sqrtf(__shfl(ssrow, v + vb, 32));
            s[v] = c[v] / fmaxf(qn * wn, EPS);
        }
        if (row == 0) {  // lanes 0 and 16 write 8 contiguous sims each
            v4f lo = {s[0], s[1], s[2], s[3]};
            v4f hi = {s[4], s[5], s[6], s[7]};
            float* out = sims + (size_t)tile * TILE_ROWS + vb;
            __builtin_nontemporal_store(lo, (v4f*)out);
            __builtin_nontemporal_store(hi, (v4f*)(out + 4));
        }
    }
}

// ---------------------------------------------------------------------------
// Top-5 helpers
// ---------------------------------------------------------------------------
__device__ __forceinline__ void top5_insert(float (&tv)[5], int (&ti)[5], float v, int i) {
    if (v <= tv[4]) return;
    int p = 4;
    while (p > 0 && v > tv[p - 1]) {
        tv[p] = tv[p - 1];
        ti[p] = ti[p - 1];
        --p;
    }
    tv[p] = v;
    ti[p] = i;
}

// Kernel 3a: per-block top-5 candidates
__global__ __launch_bounds__(256) void block_top5(const float* __restrict__ sims, int V,
                                                  float* __restrict__ cv,
                                                  int* __restrict__ ci) {
    __shared__ float sv[256 * 5];
    __shared__ int   si[256 * 5];
    float tv[5] = {-1e30f, -1e30f, -1e30f, -1e30f, -1e30f};
    int   ti[5] = {-1, -1, -1, -1, -1};
    for (int i = blockIdx.x * 256 + threadIdx.x; i < V; i += gridDim.x * 256) {
        const float v = __builtin_nontemporal_load(sims + i);
        top5_insert(tv, ti, v, i);
    }
#pragma unroll
    for (int k = 0; k < 5; ++k) {
        sv[threadIdx.x * 5 + k] = tv[k];
        si[threadIdx.x * 5 + k] = ti[k];
    }
    __syncthreads();
    if (threadIdx.x == 0) {
        float bv[5] = {-1e30f, -1e30f, -1e30f, -1e30f, -1e30f};
        int   bi[5] = {-1, -1, -1, -1, -1};
        for (int e = 0; e < 256 * 5; ++e) top5_insert(bv, bi, sv[e], si[e]);
#pragma unroll
        for (int k = 0; k < 5; ++k) {
            cv[blockIdx.x * 5 + k] = bv[k];
            ci[blockIdx.x * 5 + k] = bi[k];
        }
    }
}

// Kernel 3b: final merge, write vals then indices (as float) to d_out
__global__ void final_top5(const float* __restrict__ cv, const int* __restrict__ ci,
                           int n, float* __restrict__ out) {
    if (threadIdx.x == 0 && blockIdx.x == 0) {
        float bv[5] = {-1e30f, -1e30f, -1e30f, -1e30f, -1e30f};
        int   bi[5] = {-1, -1, -1, -1, -1};
        for (int e = 0; e < n; ++e) top5_insert(bv, bi, cv[e], ci[e]);
#pragma unroll
        for (int k = 0; k < 5; ++k) {
            out[k]     = bv[k];
            out[5 + k] = (float)bi[k];
        }
    }
}

// ---------------------------------------------------------------------------
// Launch
// ---------------------------------------------------------------------------
extern "C" void kernel_launch(void* const* d_in, const int* in_sizes, int n_in,
                              void* d_out, int out_size, void* d_ws, size_t ws_size,
                              hipStream_t stream) {
    const float* W    = (const float*)d_in[0];
    // query_idx is int64 in the reference; low 32 bits are the value (LE, V < 2^24)
    const int*   qidx = (const int*)d_in[1];
    const int    V    = in_sizes[0] / DIM;

    // Workspace layout (256B-aligned chunks)
    float* qnorm = (float*)d_ws;                       // 1 float
    float* sims  = (float*)d_ws + 64;                  // V floats
    float* cv    = sims + V;                           // TOP_BLOCKS*5 floats
    int*   ci    = (int*)(cv + TOP_BLOCKS * 5);        // TOP_BLOCKS*5 ints
    float* out   = (float*)d_out;

    qnorm_kernel<<<1, 32, 0, stream>>>(W, qidx, qnorm);
    sim_kernel<<<SIM_BLOCKS, WAVES_PER_BLOCK * 32, 0, stream>>>(W, qidx, qnorm, sims, V);
    block_top5<<<TOP_BLOCKS, 256, 0, stream>>>(sims, V, cv, ci);
    final_top5<<<1, 32, 0, stream>>>(cv, ci, TOP_BLOCKS * 5, out);
}